// MCF_3882650435661
// MI455X (gfx1250) — compile-verified
//
#include <hip/hip_runtime.h>

// Circular self-convolution via V_WMMA_F32_16X16X4_F32 (full fp32 precision).
// out[b,k] = sum_j x[b,j] * x[b,(k-j) mod N]
//
// Tile mapping (per batch b, per wave w in 0..3):
//   D[m,n] = out[b, 256*w + 16*m + n]
//          = sum_{j} x[(j + 16*m) mod N] * x[(256*w + n - j) mod N]
//   A[m,j] = x[(j + 16*m) mod N]        -> read from forward doubled image (ascending)
//   B[j,n] = x[(256*w + n - j) mod N]   -> read from REVERSED doubled image (ascending):
//            rev[u] = x[(-u) mod N],  B[kk][n] = rev[N + j + kk - 256w - n]
// Ascending fragment reads let the compiler emit in-order ds_load pairs with no
// VGPR swap movs in front of each WMMA.

#define NN 1024
#define NB 128

typedef float v2f __attribute__((ext_vector_type(2)));
typedef float v8f __attribute__((ext_vector_type(8)));

__global__ __launch_bounds__(128) void mcf_wmma_kernel(const float* __restrict__ x0,
                                                       const float* __restrict__ x1,
                                                       const float* __restrict__ x2,
                                                       float* __restrict__ out) {
    __shared__ float fwd[2 * NN];  // fwd[i] = x[i mod N]      (8 KB)
    __shared__ float rev[2 * NN];  // rev[u] = x[(-u) mod N]   (8 KB)

    const int b     = blockIdx.x;  // batch
    const int which = blockIdx.y;  // which input tensor
    const float* __restrict__ x =
        (which == 0) ? x0 : (which == 1) ? x1 : x2;
    const float* __restrict__ xb = x + b * NN;
    float* __restrict__ ob = out + (size_t)which * (NB * NN) + (size_t)b * NN;

    const int tid = threadIdx.x;   // 0..127, 4 wave32 waves

    // Cooperative fill: 8 floats/thread; forward and reversed, each doubled.
    #pragma unroll
    for (int i = tid; i < NN; i += 128) {
        float v = xb[i];
        fwd[i]      = v;
        fwd[i + NN] = v;
        int u = (0 - i) & (NN - 1);   // (N - i) mod N, branch-free
        rev[u]      = v;
        rev[u + NN] = v;
    }
    __syncthreads();

    const int wave  = tid >> 5;        // 0..3 -> output k range [256*wave, +256)
    const int lane  = tid & 31;
    const int m16   = lane & 15;       // A: row m  /  B,D: col n
    const int khalf = lane >> 4;       // 0: k={0,1}   1: k={2,3} within each K=4 chunk

    // A fragment element i:  fwd[ j + 2*khalf + i + 16*m ]        (ascending in i,j)
    const float* Abase = fwd + 16 * m16 + 2 * khalf;
    // B fragment element i:  rev[ N + j + 2*khalf + i - 256*w - n ] (ascending in i,j)
    const float* Bbase = rev + NN + 2 * khalf - 256 * wave - m16;

    v8f acc0 = {};   // j in [0, 512)
    v8f acc1 = {};   // j in [512, 1024) -- independent chain for ILP

    #pragma unroll 8
    for (int j = 0; j < NN / 2; j += 4) {
        v2f a0, b0, a1, b1;
        a0.x = Abase[j];         a0.y = Abase[j + 1];
        b0.x = Bbase[j];         b0.y = Bbase[j + 1];
        a1.x = Abase[j + 512];   a1.y = Abase[j + 513];
        b1.x = Bbase[j + 512];   b1.y = Bbase[j + 513];

        // D = A*B + C   (8 args: neg_a, A, neg_b, B, c_mod, C, reuse_a, reuse_b)
        acc0 = __builtin_amdgcn_wmma_f32_16x16x4_f32(
            false, a0, false, b0, (short)0, acc0, false, false);
        acc1 = __builtin_amdgcn_wmma_f32_16x16x4_f32(
            false, a1, false, b1, (short)0, acc1, false, false);
    }

    v8f acc = acc0 + acc1;

    // D layout: reg r holds M = r + 8*khalf, N = lane&15.
    // out index = 256*wave + 16*M + n
    #pragma unroll
    for (int r = 0; r < 8; ++r) {
        ob[256 * wave + 16 * (r + 8 * khalf) + m16] = acc[r];
    }
}

extern "C" void kernel_launch(void* const* d_in, const int* in_sizes, int n_in,
                              void* d_out, int out_size, void* d_ws, size_t ws_size,
                              hipStream_t stream) {
    const float* x1 = (const float*)d_in[0];
    const float* x2 = (const float*)d_in[1];
    const float* x3 = (const float*)d_in[2];
    float* out = (float*)d_out;

    dim3 grid(NB, 3);   // 128 batches x 3 inputs
    dim3 block(128);    // 4 wave32 waves; one 16x16 output tile chain per wave
    mcf_wmma_kernel<<<grid, block, 0, stream>>>(x1, x2, x3, out);
}